// LR_77558519431935
// MI455X (gfx1250) — compile-verified
//
#include <hip/hip_runtime.h>
#include <hip/hip_bf16.h>

typedef float v2f __attribute__((ext_vector_type(2)));
typedef float v8f __attribute__((ext_vector_type(8)));

#define BATCH          65536
#define NSPARSE        20
#define DCOLS          256
#define STOTAL         488000
#define ROWS_PER_BLOCK 128
#define KCHUNK         32
#define NCHUNK         (DCOLS / KCHUNK)
#define KPAD           36   // stride 36 words: conflict-free b64 A-fragment reads

__constant__ int c_offs[NSPARSE] = {
    0, 100000, 200000, 300000,
    400000, 410000, 420000, 430000, 440000, 450000, 460000, 470000,
    480000, 481000, 482000, 483000, 484000, 485000, 486000, 487000
};

// ---------- CDNA5 async global->LDS copy (ASYNCcnt-tracked) ----------
#if defined(__has_builtin)
#if __has_builtin(__builtin_amdgcn_global_load_async_to_lds_b128)
#define HAVE_ASYNC_B128 1
#endif
#if __has_builtin(__builtin_amdgcn_s_wait_asynccnt)
#define HAVE_WAIT_ASYNC 1
#endif
#endif

// Builtin param 0 is 'int __vector_size__(16) __device__*'  (= addrspace(1)).
typedef int b128v __attribute__((vector_size(16)));
typedef __attribute__((address_space(1))) b128v gb128;
typedef __attribute__((address_space(3))) b128v lb128;
typedef __attribute__((address_space(3))) void  las_void;

__device__ __forceinline__ void async_copy16(const void* g, void* l) {
#ifdef HAVE_ASYNC_B128
    void* gnc = (void*)g;   // drop const, then addrspace-cast
    __builtin_amdgcn_global_load_async_to_lds_b128((gb128*)gnc, (lb128*)l, 0, 0);
#else
    unsigned int       loff = (unsigned int)(unsigned long long)(las_void*)l;
    unsigned long long ga   = (unsigned long long)g;
    asm volatile("global_load_async_to_lds_b128 %0, %1, off"
                 :: "v"(loff), "v"(ga) : "memory");
#endif
}

#ifdef HAVE_WAIT_ASYNC
#define WAIT_ASYNC(n) __builtin_amdgcn_s_wait_asynccnt(n)
#else
#define WAIT_ASYNC(n) asm volatile("s_wait_asynccnt " #n ::: "memory")
#endif
// ---------------------------------------------------------------------

__launch_bounds__(256, 2)
__global__ void fm_first_order_kernel(const int*   __restrict__ sparse,
                                      const float* __restrict__ dense,
                                      const float* __restrict__ W,
                                      const float* __restrict__ bias,
                                      float*       __restrict__ out) {
    // Double-buffered 128x32 f32 tile (2 x 18KB) + dense W slice (1KB)
    __shared__ __align__(16) float tile[2][ROWS_PER_BLOCK][KPAD];
    __shared__ float sw[DCOLS];

    const int tid  = threadIdx.x;
    const int lane = tid & 31;
    const int wave = tid >> 5;      // 0..7, 16 rows each
    const int half = lane >> 4;     // K pairs 0,1 vs 2,3
    const int l16  = lane & 15;

    const int rowBase = blockIdx.x * ROWS_PER_BLOCK;

    sw[tid] = W[STOTAL + tid];      // stage dense slice of W

    const int loadRow = tid >> 3;        // 0..31, 4 rows per thread
    const int loadCol = (tid & 7) * 4;   // float4 column within chunk

    // Issue all 4 async 16B copies of one chunk into one LDS buffer.
    auto issue_chunk = [&](int c, int buf) {
        #pragma unroll
        for (int rr = 0; rr < 4; ++rr) {
            const int r = rr * 32 + loadRow;
            const float* g = dense + (size_t)(rowBase + r) * DCOLS
                                   + c * KCHUNK + loadCol;
            async_copy16(g, &tile[buf][r][loadCol]);
        }
    };

    v8f acc = {};   // 16x16 f32 C/D accumulator (8 VGPRs)

    issue_chunk(0, 0);  // prologue

    for (int c = 0; c < NCHUNK; ++c) {
        if (c + 1 < NCHUNK) {
            issue_chunk(c + 1, (c + 1) & 1);   // prefetch next chunk
            WAIT_ASYNC(4);   // in-order: chunk c's 4 copies complete
        } else {
            WAIT_ASYNC(0);
        }
        __syncthreads();     // whole tile for chunk c visible to all waves

        const float (*tb)[KPAD] = tile[c & 1];
        const int wrb = wave * 16;
        #pragma unroll
        for (int k = 0; k < KCHUNK; k += 4) {
            v2f a, b;
            // A (16x4): lanes 0-15 hold K=0,1; lanes 16-31 hold K=2,3 of row l16
            a.x = tb[wrb + l16][k + 2 * half];
            a.y = tb[wrb + l16][k + 2 * half + 1];
            // B (4x16): w[k] broadcast across all 16 N columns
            const int kg = c * KCHUNK + k;
            b.x = sw[kg + 2 * half];
            b.y = sw[kg + 2 * half + 1];
            acc = __builtin_amdgcn_wmma_f32_16x16x4_f32(
                      false, a, false, b, (short)0, acc, false, false);
        }
        __syncthreads();     // all waves done reading buf (c&1) before reuse
    }

    // D[m,n] replicated over n. Lanes 0-7 own rows 0-7 (acc[m]),
    // lanes 16-23 own rows 8-15 (acc[m-8]).
    if (l16 < 8) {
        float dres = 0.f;
        #pragma unroll
        for (int i = 0; i < 8; ++i) dres = (l16 == i) ? acc[i] : dres;

        const int row = rowBase + wave * 16 + half * 8 + l16;

        float ssum = 0.f;
        #pragma unroll
        for (int j = 0; j < NSPARSE; ++j) {
            const int id = sparse[(size_t)row * NSPARSE + j];
            ssum += W[c_offs[j] + id];
        }

        const float logit = dres + ssum + bias[0];
        out[row] = 1.0f / (1.0f + __expf(-logit));
    }
}

extern "C" void kernel_launch(void* const* d_in, const int* in_sizes, int n_in,
                              void* d_out, int out_size, void* d_ws, size_t ws_size,
                              hipStream_t stream) {
    (void)in_sizes; (void)n_in; (void)d_ws; (void)ws_size; (void)out_size;
    const int*   sparse = (const int*)  d_in[0];
    const float* dense  = (const float*)d_in[1];
    const float* W      = (const float*)d_in[2];
    const float* bias   = (const float*)d_in[3];
    float* out = (float*)d_out;

    dim3 grid(BATCH / ROWS_PER_BLOCK);
    fm_first_order_kernel<<<grid, dim3(256), 0, stream>>>(sparse, dense, W, bias, out);
}